// ManyBodyVoxel_9715216023993
// MI455X (gfx1250) — compile-verified
//
#include <hip/hip_runtime.h>
#include <hip/hip_bf16.h>

// Voxelized Gaussian density on MI455X (gfx1250, wave32).
//
// Formulation: for each (b,a,type z,grid-length l) channel,
//   out[g] = sum_{n: an[a,n]==type[z]} exp(coeff * (|g|^2 - 2 g.d_n + |d_n|^2))
// The -2 g.d + |d|^2 part is a K=4 f32 matmul -> V_WMMA_F32_16X16X4_F32:
//   A (16 neighbors x 4) row = (-2c*dx, -2c*dy, -2c*dz, c*|d|^2)
//   B (4 x 16 grid pts) col  = (gx, gy, gz, 1)
// then add c*|g|^2 per column, v_exp_f32, and reduce rows.
// A small pre-pass buckets neighbors per (atom, type) so the one-hot masked
// reduction disappears; padded A rows carry -1e30 in the |d|^2 slot so their
// exp underflows to exactly 0.

typedef __attribute__((ext_vector_type(2))) float v2f;
typedef __attribute__((ext_vector_type(8))) float v8f;

#define NB 2      // batch
#define NA 64     // atoms
#define NN 48     // neighbors
#define NZ 4      // atom types
#define NL 2      // grid lengths
#define G  16     // grid size per dim
#define G3 4096   // 16^3

// ---------------- pre-pass: bucket neighbors per (atom, type) ----------------
__global__ void build_lists_kernel(const int* __restrict__ an,
                                   int* __restrict__ cnt,
                                   int* __restrict__ idx) {
    int a = threadIdx.x;
    if (a >= NA) return;
    const int types[NZ] = {1, 6, 7, 8};
    int c[NZ] = {0, 0, 0, 0};
    for (int n = 0; n < NN; ++n) {
        int v = an[a * NN + n];
        #pragma unroll
        for (int z = 0; z < NZ; ++z) {
            if (v == types[z]) {
                idx[(a * NZ + z) * NN + c[z]] = n;
                c[z]++;
            }
        }
    }
    #pragma unroll
    for (int z = 0; z < NZ; ++z) cnt[a * NZ + z] = c[z];
}

// ---------------- main kernel: WMMA gaussian accumulation --------------------
// wave decomposition: 8192 waves = b(2) * a(64) * z(4) * chunk(16);
// each wave handles 16 column-tiles of 16 grid points, for both grid lengths.
__global__ void __launch_bounds__(256)
voxel_wmma_kernel(const float* __restrict__ dv,     // (2,64,48,3)
                  const float* __restrict__ sigma_p,
                  const int* __restrict__ cnt,      // (64,4)
                  const int* __restrict__ idxl,     // (64,4,48)
                  float* __restrict__ out) {        // (2,64,8,16,16,16)
    const int lane = threadIdx.x & 31;
    const int half = lane >> 4;        // 0: lanes 0-15, 1: lanes 16-31
    const int row  = lane & 15;        // M index for A, N index for B/D

    const int w     = (blockIdx.x * blockDim.x + threadIdx.x) >> 5;
    const int chunk = w & 15;          // 16 column-tiles per chunk
    const int z     = (w >> 4) & 3;
    const int a     = (w >> 6) & 63;
    const int b     = (w >> 12) & 1;

    const float sigma = sigma_p[0];
    const float coeff = -0.5f / (sigma * sigma);
    const float c2    = -2.0f * coeff;

    const int count = cnt[a * NZ + z];
    float* outc = out + (size_t)(((b * NA + a) * (NZ * NL) + z * NL)) * G3;

    if (count == 0) {
        // both length-channels of this (b,a,z) chunk are zero
        if (half == 0) {
            for (int l = 0; l < NL; ++l)
                for (int ct = 0; ct < 16; ++ct)
                    outc[l * G3 + (chunk * 16 + ct) * 16 + row] = 0.0f;
        }
        return;
    }

    const int ntiles = (count + 15) >> 4;   // 1..3

    // Preload A fragments for up to 3 K-accumulation tiles.
    // A 16x4 f32 layout: VGPR0 = K0 (lanes 0-15) | K2 (lanes 16-31),
    //                    VGPR1 = K1 (lanes 0-15) | K3 (lanes 16-31).
    float a0[3], a1[3];
    #pragma unroll
    for (int t = 0; t < 3; ++t) {
        float r0 = 0.0f, r1 = 0.0f, r2 = 0.0f, r3 = -1e30f;  // pad row
        const int pos = t * 16 + row;
        if (t < ntiles && pos < count) {
            const int n = idxl[(a * NZ + z) * NN + pos];
            const float* dp = dv + (size_t)(((b * NA + a) * NN + n)) * 3;
            const float dx = dp[0], dy = dp[1], dz = dp[2];
            r0 = c2 * dx;
            r1 = c2 * dy;
            r2 = c2 * dz;
            r3 = coeff * (dx * dx + dy * dy + dz * dz);
        }
        a0[t] = half ? r2 : r0;
        a1[t] = half ? r3 : r1;
    }

    for (int l = 0; l < NL; ++l) {
        const float glen  = l ? 6.0f : 3.0f;
        const float step  = glen / 15.0f;
        const float halfg = 0.5f * glen;

        for (int ct = 0; ct < 16; ++ct) {
            // column N = row; grid flat index for this lane's column
            const int g  = (chunk * 16 + ct) * 16 + row;
            const int xi = g >> 8, yi = (g >> 4) & 15, zi = g & 15;
            const float gx = xi * step - halfg;
            const float gy = yi * step - halfg;
            const float gz = zi * step - halfg;
            const float tcol = coeff * (gx * gx + gy * gy + gz * gz);

            // B 4x16 f32 layout: VGPR0 = K0 | K2, VGPR1 = K1 | K3 (by half)
            v2f bfrag;
            bfrag[0] = half ? gz : gx;
            bfrag[1] = half ? 1.0f : gy;

            float acc = 0.0f;
            #pragma unroll
            for (int t = 0; t < 3; ++t) {
                if (t < ntiles) {
                    v2f afrag;
                    afrag[0] = a0[t];
                    afrag[1] = a1[t];
                    v8f c = {};
                    // D[M][N] = coeff*(-2 g.d + |d|^2); M = v + 8*half, N = row
                    v8f d = __builtin_amdgcn_wmma_f32_16x16x4_f32(
                        false, afrag, false, bfrag, (short)0, c, false, false);
                    #pragma unroll
                    for (int v = 0; v < 8; ++v)
                        acc += __expf(d[v] + tcol);   // exp(coeff*||g-d||^2)
                }
            }
            // combine rows 0-7 (half 0) with rows 8-15 (half 1)
            acc += __shfl_xor(acc, 16, 32);
            if (half == 0)
                outc[l * G3 + (chunk * 16 + ct) * 16 + row] = acc;
        }
    }
}

extern "C" void kernel_launch(void* const* d_in, const int* in_sizes, int n_in,
                              void* d_out, int out_size, void* d_ws, size_t ws_size,
                              hipStream_t stream) {
    const float* dv    = (const float*)d_in[0];   // distance_vector (2,64,48,3)
    const float* sigma = (const float*)d_in[1];   // sigma (1,)
    const int*   an    = (const int*)d_in[2];     // atomic_numbers (64,48)
    float* out = (float*)d_out;                   // (2,64,8,16,16,16) f32

    int* cnt = (int*)d_ws;                         // 64*4 ints
    int* idx = (int*)((char*)d_ws + 1024);         // 64*4*48 ints

    build_lists_kernel<<<1, 64, 0, stream>>>(an, cnt, idx);

    // 8192 waves = 2*64*4*16 ; 256 threads/block (8 waves) -> 1024 blocks
    voxel_wmma_kernel<<<1024, 256, 0, stream>>>(dv, sigma, cnt, idx, out);
}